// AttentionLayer_23244363006604
// MI455X (gfx1250) — compile-verified
//
#include <hip/hip_runtime.h>
#include <hip/hip_bf16.h>

typedef __attribute__((ext_vector_type(2))) float v2f;
typedef __attribute__((ext_vector_type(8))) float v8f;
typedef int v4i __attribute__((vector_size(16)));   // matches builtin prototype

#define E_DIM 512
#define K_SEL 36

#define HAVE_ASYNC_LDS (__has_builtin(__builtin_amdgcn_global_load_async_to_lds_b128) && \
                        __has_builtin(__builtin_amdgcn_s_wait_asynccnt))

// ---------------------------------------------------------------------------
// Kernel 1: v[e] = sum_f W1[f,e] * W2[0, E + f]   (v = W1^T * w2y)
// Done with V_WMMA_F32_16X16X4_F32: one wave per 16 rows of v.
// A tile (16x4):  A[m][k] = W1[(f0+k)*E + e0+m]
// B tile (4x16):  column 0 = w2y[f0 .. f0+3], other columns zero
// D column 0 accumulates v[e0 .. e0+15].
// VGPR layouts per CDNA5 ISA 7.12.2 (32-bit A 16x4; 32-bit C/D 16x16).
// ---------------------------------------------------------------------------
__global__ __launch_bounds__(32) void build_v_wmma(const float* __restrict__ W1,
                                                   const float* __restrict__ W2,
                                                   float* __restrict__ v_out) {
  const int lane = threadIdx.x;          // 0..31, EXEC all ones at WMMA
  const int M    = lane & 15;            // row (A) / column (B) index
  const int k0   = (lane >> 4) << 1;     // lanes 0-15: K=0,1 ; lanes 16-31: K=2,3
  const int e0   = blockIdx.x * 16;
  const float* __restrict__ w2y = W2 + E_DIM;

  v8f acc = {};
#pragma unroll 4
  for (int f0 = 0; f0 < E_DIM; f0 += 4) {
    v2f a, b;
    a.x = W1[(size_t)(f0 + k0)     * E_DIM + e0 + M];
    a.y = W1[(size_t)(f0 + k0 + 1) * E_DIM + e0 + M];
    b.x = (M == 0) ? w2y[f0 + k0]     : 0.0f;
    b.y = (M == 0) ? w2y[f0 + k0 + 1] : 0.0f;
    acc = __builtin_amdgcn_wmma_f32_16x16x4_f32(
        /*neg_a=*/false, a, /*neg_b=*/false, b,
        /*c_mod=*/(short)0, acc, /*reuse_a=*/false, /*reuse_b=*/false);
  }
  // D layout: VGPR j -> lanes 0-15: M=j, N=lane ; lanes 16-31: M=8+j, N=lane-16.
  // Column N==0 lives in lane 0 (M=j) and lane 16 (M=8+j).
  if (lane == 0) {
#pragma unroll
    for (int j = 0; j < 8; ++j) v_out[e0 + j] = acc[j];
  } else if (lane == 16) {
#pragma unroll
    for (int j = 0; j < 8; ++j) v_out[e0 + 8 + j] = acc[j];
  }
}

// ---------------------------------------------------------------------------
// Kernel 2: fused logits + softmax + weighted sum, one workgroup per n.
// y[n] (36x512 f32 = 72 KB) is staged to LDS once (async-to-LDS when
// available), so HBM traffic is y read exactly once + output write.
// ---------------------------------------------------------------------------
__global__ __launch_bounds__(256) void attn_fused(const float* __restrict__ y,
                                                  const float* __restrict__ v,
                                                  float* __restrict__ out) {
  __shared__ __align__(16) float sY[K_SEL * E_DIM];   // 73728 B
  __shared__ __align__(16) float sV[E_DIM];
  __shared__ float sLogit[K_SEL];
  __shared__ float sExp[K_SEL];

  const int tid = threadIdx.x;
  const int n   = blockIdx.x;
  const float* __restrict__ yb = y + (size_t)n * (K_SEL * E_DIM);

  // stage v (2 KB) with plain vector loads
  if (tid < E_DIM / 4) {
    ((float4*)sV)[tid] = ((const float4*)v)[tid];
  }

  // stage y[n] tile: 4608 x 16B, 18 issues per thread
#if HAVE_ASYNC_LDS
  {
    const char* g = (const char*)yb;
    char*       l = (char*)sY;
#pragma unroll
    for (int i = 0; i < (K_SEL * E_DIM) / (4 * 256); ++i) {
      const int off = (tid + i * 256) * 16;
      __builtin_amdgcn_global_load_async_to_lds_b128(
          (__attribute__((address_space(1))) v4i*)(g + off),
          (__attribute__((address_space(3))) v4i*)(l + off),
          0, 0);
    }
    __builtin_amdgcn_s_wait_asynccnt(0);
  }
#else
#pragma unroll
  for (int i = 0; i < (K_SEL * E_DIM) / (4 * 256); ++i) {
    const int idx = tid + i * 256;
    ((float4*)sY)[idx] = ((const float4*)yb)[idx];
  }
#endif
  __syncthreads();

  // Phase 1: logits s[k] = y[n,k,:] . v  -- one wave per k, strided over 8 waves
  const int wave = tid >> 5;
  const int lane = tid & 31;
  for (int k = wave; k < K_SEL; k += 8) {
    const float* __restrict__ row = sY + k * E_DIM;
    float p = 0.0f;
#pragma unroll
    for (int j = 0; j < E_DIM / 32; ++j)
      p += row[lane + j * 32] * sV[lane + j * 32];
#pragma unroll
    for (int off = 16; off > 0; off >>= 1)
      p += __shfl_xor(p, off, 32);
    if (lane == 0) sLogit[k] = p;
  }
  __syncthreads();

  // Phase 2: numerically stable exp
  if (tid < K_SEL) {
    float mx = -INFINITY;
#pragma unroll
    for (int k = 0; k < K_SEL; ++k) mx = fmaxf(mx, sLogit[k]);
    sExp[tid] = __expf(sLogit[tid] - mx);
  }
  __syncthreads();

  // Phase 3: a[n,e] = (1/sum) * sum_k exp[k] * y[n,k,e]; each thread owns 2 e's
  float sum = 0.0f;
#pragma unroll
  for (int k = 0; k < K_SEL; ++k) sum += sExp[k];
  const float inv = 1.0f / sum;

  const int e0 = tid;
  const int e1 = tid + 256;
  float a0 = 0.0f, a1 = 0.0f;
#pragma unroll 4
  for (int k = 0; k < K_SEL; ++k) {
    const float w = sExp[k];
    a0 += w * sY[k * E_DIM + e0];
    a1 += w * sY[k * E_DIM + e1];
  }
  out[(size_t)n * E_DIM + e0] = a0 * inv;
  out[(size_t)n * E_DIM + e1] = a1 * inv;
}

// ---------------------------------------------------------------------------
// inputs: 0:x 1:y 2:W1 3:b1 4:W2 5:b2 6:K   (x, b1, b2 cancel in softmax)
// ---------------------------------------------------------------------------
extern "C" void kernel_launch(void* const* d_in, const int* in_sizes, int n_in,
                              void* d_out, int out_size, void* d_ws, size_t ws_size,
                              hipStream_t stream) {
  const float* y  = (const float*)d_in[1];
  const float* W1 = (const float*)d_in[2];
  const float* W2 = (const float*)d_in[4];
  float* v   = (float*)d_ws;   // 512 floats of scratch, fully overwritten
  float* out = (float*)d_out;

  const int N = in_sizes[1] / (K_SEL * E_DIM);   // B*T = 2048

  build_v_wmma<<<E_DIM / 16, 32, 0, stream>>>(W1, W2, v);
  attn_fused<<<N, 256, 0, stream>>>(y, v, out);
}